// RzLinear_80771154968693
// MI455X (gfx1250) — compile-verified
//
#include <hip/hip_runtime.h>

typedef __attribute__((ext_vector_type(2))) float v2f;
typedef __attribute__((ext_vector_type(4))) float v4f;
typedef __attribute__((ext_vector_type(8))) float v8f;
typedef __attribute__((__vector_size__(4 * sizeof(int)))) int v4i;

#define GAS __attribute__((address_space(1)))
#define LAS __attribute__((address_space(3)))

#define BKT 32            // hashed tile K dim
#define BLOCK_M 64        // = batch B
#define BLOCK_N 64        // two hashed tiles wide
#define PX 36             // x-tile pitch (floats): 144B rows, 16B aligned, conflict-free A reads
#define PW 34             // W-tile pitch (floats): 136B rows, 8B aligned, conflict-free scatter+reads
#define P_MOD 2038074743u // ROBE-Z hash modulus

#if __has_builtin(__builtin_amdgcn_global_load_async_to_lds_b32) && \
    __has_builtin(__builtin_amdgcn_global_load_async_to_lds_b128)
#define USE_ASYNC 1
#else
#define USE_ASYNC 0
#endif

__device__ __forceinline__ void wait_async_zero() {
#if USE_ASYNC
#if __has_builtin(__builtin_amdgcn_s_wait_asynccnt)
    __builtin_amdgcn_s_wait_asynccnt(0);
#else
    asm volatile("s_wait_asynccnt 0" ::: "memory");
#endif
#endif
}

__global__ __launch_bounds__(256)
void rz_linear_wmma_f32(const float* __restrict__ x,
                        const float* __restrict__ hw,
                        const int*   __restrict__ rn,
                        const float* __restrict__ bias,
                        float*       __restrict__ out,
                        int K, int N, unsigned int Hmod)
{
    __shared__ float xs[2][BLOCK_M * PX];   // x tile ping-pong: [m][k_local]
    __shared__ float wt[2][BLOCK_N * PW];   // W tile transposed ping-pong: [n_local][k_local]

    const int tid  = threadIdx.x;
    const int wave = tid >> 5;
    const int lane = tid & 31;
    const int hl   = lane >> 4;             // K-pair select per ISA 16x4 A/B layout
    const int l16  = lane & 15;
    const int wm   = wave & 3;              // M tile id (0..3)
    const int wn   = wave >> 2;             // N half id (0..1)

    const unsigned int R1 = (unsigned int)rn[1];
    const unsigned int R2 = (unsigned int)rn[2];
    const unsigned int R3 = (unsigned int)rn[3];

    const int tn0     = blockIdx.x * 2;
    const int n_block = blockIdx.x * BLOCK_N;

    // h_i = (R1 + tn_i*R2 + kt*R3) mod P, advanced by conditional subtract (no div in loop)
    unsigned int h0 = (unsigned int)((((unsigned long long)tn0       * R2) + R1) % P_MOD);
    unsigned int h1 = (unsigned int)((((unsigned long long)(tn0 + 1) * R2) + R1) % P_MOD);
    const unsigned int r3m = R3 % P_MOD;

    // Stage one K-tile (x: 64x32, W: 2 hashed 32x32 tiles stored transposed).
    auto stage = [&](float* xbuf, float* wbuf, int k0,
                     unsigned int s0, unsigned int s1) {
#if USE_ASYNC
        {   // x tile: 512 float4, 2 per thread, async b128 (global coalesced, LDS 16B aligned)
            const int m8  = tid >> 3;
            const int kk4 = (tid & 7) << 2;
            #pragma unroll
            for (int i = 0; i < 2; ++i) {
                const int m = m8 + (i << 5);
                __builtin_amdgcn_global_load_async_to_lds_b128(
                    (GAS v4i*)(x + (size_t)m * K + k0 + kk4),
                    (LAS v4i*)(xbuf + m * PX + kk4), 0, 0);
            }
        }
        {   // weight tiles: transpose happens in the async scatter addressing (conflict-free @ PW=34)
            const int r = tid >> 3;             // k row in tile (0..31)
            const int c = (tid & 7) << 2;       // n col base (0,4,..,28)
            #pragma unroll
            for (int t = 0; t < 2; ++t) {       // two hashed tiles
                const unsigned int st = t ? s1 : s0;
                const float* src = hw + st + r * 32 + c;
                float*       dst = wbuf + (t * 32 + c) * PW + r;
                #pragma unroll
                for (int j = 0; j < 4; ++j) {
                    __builtin_amdgcn_global_load_async_to_lds_b32(
                        (GAS int*)(src + j),
                        (LAS int*)(dst + j * PW), 0, 0);
                }
            }
        }
#else
        {   // fallback: loads through VGPRs + ds stores
            const int m8  = tid >> 3;
            const int kk4 = (tid & 7) << 2;
            #pragma unroll
            for (int i = 0; i < 2; ++i) {
                const int m = m8 + (i << 5);
                v4f v = *(const v4f*)(x + (size_t)m * K + k0 + kk4);
                *(v4f*)&xbuf[m * PX + kk4] = v;
            }
        }
        {
            const int r = tid >> 3;
            const int c = (tid & 7) << 2;
            #pragma unroll
            for (int t = 0; t < 2; ++t) {
                const unsigned int st = t ? s1 : s0;
                const float* src = hw + st + r * 32 + c;   // 4B-aligned only
                const float w0 = src[0], w1 = src[1], w2 = src[2], w3 = src[3];
                float* dst = wbuf + (t * 32 + c) * PW + r;
                dst[0 * PW] = w0; dst[1 * PW] = w1; dst[2 * PW] = w2; dst[3 * PW] = w3;
            }
        }
#endif
    };

    // C = broadcast bias, so bias is free in the accumulator init.
    const int nb0 = n_block + wn * 32 + l16;
    const float bv0 = bias[nb0];
    const float bv1 = bias[nb0 + 16];
    v8f acc0 = {bv0, bv0, bv0, bv0, bv0, bv0, bv0, bv0};
    v8f acc1 = {bv1, bv1, bv1, bv1, bv1, bv1, bv1, bv1};

    const int ktiles = K / BKT;

    // prologue: stage tile 0 into buffer 0
    stage(xs[0], wt[0], 0, h0 % Hmod, h1 % Hmod);
    h0 += r3m; if (h0 >= P_MOD) h0 -= P_MOD;
    h1 += r3m; if (h1 >= P_MOD) h1 -= P_MOD;

    for (int kt = 0; kt < ktiles; ++kt) {
        wait_async_zero();       // my async portion of tile kt complete
        __syncthreads();         // everyone's portion complete; everyone done reading buf[kt^1]

        const int cur = kt & 1;
        if (kt + 1 < ktiles) {   // kick off tile kt+1 into the other buffer (overlaps compute)
            stage(xs[cur ^ 1], wt[cur ^ 1], (kt + 1) * BKT, h0 % Hmod, h1 % Hmod);
            h0 += r3m; if (h0 >= P_MOD) h0 -= P_MOD;
            h1 += r3m; if (h1 >= P_MOD) h1 -= P_MOD;
        }

        // 8 K-steps of v_wmma_f32_16x16x4_f32, 2 N-tiles per wave
        const float* xrow  = &xs[cur][(wm * 16 + l16) * PX];
        const float* w0row = &wt[cur][(wn * 32 +      l16) * PW];
        const float* w1row = &wt[cur][(wn * 32 + 16 + l16) * PW];
        #pragma unroll
        for (int s = 0; s < 8; ++s) {
            const int o = s * 4 + 2 * hl;
            v2f a  = *(const v2f*)(xrow  + o);
            v2f b0 = *(const v2f*)(w0row + o);
            v2f b1 = *(const v2f*)(w1row + o);
            acc0 = __builtin_amdgcn_wmma_f32_16x16x4_f32(
                       false, a, false, b0, (short)0, acc0, false, false);
            acc1 = __builtin_amdgcn_wmma_f32_16x16x4_f32(
                       false, a, false, b1, (short)0, acc1, false, false);
        }
    }

    // D layout: VGPR j -> M = j + 8*hl, N = l16 (bias already accumulated via C)
    #pragma unroll
    for (int nt = 0; nt < 2; ++nt) {
        const v8f a = nt ? acc1 : acc0;
        const int n = n_block + wn * 32 + nt * 16 + l16;
        #pragma unroll
        for (int j = 0; j < 8; ++j) {
            out[(size_t)(wm * 16 + j + 8 * hl) * N + n] = a[j];
        }
    }
}

extern "C" void kernel_launch(void* const* d_in, const int* in_sizes, int n_in,
                              void* d_out, int out_size, void* d_ws, size_t ws_size,
                              hipStream_t stream) {
    const float* x    = (const float*)d_in[0];
    const float* hw   = (const float*)d_in[1];
    const int*   rn   = (const int*)d_in[2];
    const float* bias = (const float*)d_in[3];
    float*       out  = (float*)d_out;

    const int N  = in_sizes[3];                // 8192 (bias length)
    const int Bm = out_size / N;               // 64   (batch)
    const int K  = in_sizes[0] / Bm;           // 4096
    const unsigned int Hmod = (unsigned int)(in_sizes[1] - BKT * 32);  // H - 1024

    dim3 grid(N / BLOCK_N);                    // 128 blocks
    rz_linear_wmma_f32<<<grid, 256, 0, stream>>>(x, hw, rn, bias, out, K, N, Hmod);
}